// ST_GCNN_layer_20057497272943
// MI455X (gfx1250) — compile-verified
//
#include <hip/hip_runtime.h>

// CDNA5 / gfx1250. fp32 end-to-end via V_WMMA_F32_16X16X4_F32.
typedef __attribute__((ext_vector_type(2))) float v2f;
typedef __attribute__((ext_vector_type(8))) float v8f;

#define N_DIM   256
#define C_DIM   64
#define T_DIM   64
#define V_DIM   22
#define NC_TILE 16
#define XS_STR  68            // padded t-stride in LDS: bank = (4m + t) % 64, conflict-free
#define ROW_LEN (T_DIM * V_DIM)   // 1408 floats per (n,c) plane

// ---------------------------------------------------------------------------
// Kernel 1: fused temporal einsum (WMMA f32 16x16x4) + spatial einsum (VALU).
//   x1[m,q,v] = sum_t x[m,t,v] * Tp[v,t,q]        (per-v 16x64x64 GEMM, WMMA)
//   x2[m,t,w] = sum_v x1[m,t,v] * (A[t]+Afix)[v,w] (K=22, VALU)
// Each block owns 16 consecutive (n*C+c) rows.
// ---------------------------------------------------------------------------
__global__ __launch_bounds__(256) void stgcn_graph_kernel(
    const float* __restrict__ x, const float* __restrict__ Tp,
    const float* __restrict__ A, const float* __restrict__ Afix,
    float* __restrict__ x2out)
{
  extern __shared__ float xs[];                // [22][16][68] = 95,744 B
  const int tid  = threadIdx.x;
  const int lane = tid & 31;
  const int wave = tid >> 5;                   // 8 waves (wave32)
  const int hi   = (lane >> 4) & 1;            // lane half: selects K pair
  const int l15  = lane & 15;
  const int ncBase = blockIdx.x * NC_TILE;

  // ---- stage x tile, transposed to xs[v][m][t]; global reads fully coalesced
  const float* xblk = x + (size_t)ncBase * ROW_LEN;
  for (int idx = tid; idx < NC_TILE * ROW_LEN; idx += 256) {
    int m   = idx / ROW_LEN;
    int rem = idx - m * ROW_LEN;
    int t   = rem / V_DIM;
    int v   = rem - t * V_DIM;
    xs[(v * NC_TILE + m) * XS_STR + t] = xblk[idx];
  }
  __syncthreads();

  // ---- phase 1: per-v temporal GEMM, x1 overwrites xs[v] in place
  for (int v = wave; v < V_DIM; v += 8) {
    v8f acc0 = {}, acc1 = {}, acc2 = {}, acc3 = {};
    const float* tpv  = Tp + (size_t)v * T_DIM * T_DIM;     // Tp[v][t][q]
    const float* xrow = &xs[(v * NC_TILE + l15) * XS_STR];  // A-frag row (M = l15)
#pragma unroll 4
    for (int k = 0; k < 16; ++k) {
      const int col = 4 * k + (hi ? 2 : 0);   // K index pair for this lane half
      v2f a;                                   // A 16x4: lane M=l15, K = col, col+1
      a.x = xrow[col];
      a.y = xrow[col + 1];
      const float* bp = tpv + (size_t)col * T_DIM + l15;    // B 4x16: K = col/col+1, N = q
      v2f b;
      b.x = bp[0];            b.y = bp[T_DIM];
      acc0 = __builtin_amdgcn_wmma_f32_16x16x4_f32(false, a, false, b, (short)0, acc0, false, false);
      b.x = bp[16];           b.y = bp[T_DIM + 16];
      acc1 = __builtin_amdgcn_wmma_f32_16x16x4_f32(false, a, false, b, (short)0, acc1, false, false);
      b.x = bp[32];           b.y = bp[T_DIM + 32];
      acc2 = __builtin_amdgcn_wmma_f32_16x16x4_f32(false, a, false, b, (short)0, acc2, false, false);
      b.x = bp[48];           b.y = bp[T_DIM + 48];
      acc3 = __builtin_amdgcn_wmma_f32_16x16x4_f32(false, a, false, b, (short)0, acc3, false, false);
    }
    // C/D layout: VGPR r -> M = r + 8*hi, N = l15 (+ q-tile offset)
#pragma unroll
    for (int r = 0; r < 8; ++r) {
      const int m = r + (hi << 3);
      float* dst = &xs[(v * NC_TILE + m) * XS_STR];
      dst[ 0 + l15] = acc0[r];
      dst[16 + l15] = acc1[r];
      dst[32 + l15] = acc2[r];
      dst[48 + l15] = acc3[r];
    }
  }
  __syncthreads();

  // ---- phase 2: spatial einsum, K = V = 22 (VALU; shape unfit for 16x16 WMMA)
  for (int p = tid; p < NC_TILE * T_DIM; p += 256) {
    const int m = p >> 6;
    const int t = p & 63;
    float r1[V_DIM];
#pragma unroll
    for (int v = 0; v < V_DIM; ++v) r1[v] = xs[(v * NC_TILE + m) * XS_STR + t];
    float ow[V_DIM];
#pragma unroll
    for (int w = 0; w < V_DIM; ++w) ow[w] = 0.f;
    const float* At = A + (size_t)t * V_DIM * V_DIM;
    for (int v = 0; v < V_DIM; ++v) {
      const float xv = r1[v];
      const float* ar = At   + v * V_DIM;
      const float* fr = Afix + v * V_DIM;
#pragma unroll
      for (int w = 0; w < V_DIM; ++w) ow[w] = fmaf(xv, ar[w] + fr[w], ow[w]);
    }
    float* dst = x2out + (size_t)(ncBase + m) * ROW_LEN + t * V_DIM;
#pragma unroll
    for (int w = 0; w < V_DIM; ++w) dst[w] = ow[w];
  }
}

// ---------------------------------------------------------------------------
// Kernel 2: 3x3 conv over (T,V) as 9 shifted GEMMs (WMMA f32) + bias + residual.
// Block = (n, 16-row t-block). LDS holds zero-padded x2 halo tile [64][18][24].
// GEMM: out[o, p] = bc[o] + x[n,o,p] + sum_{dt,dv} Wc[:,:,dt,dv] @ Xshift[:,p]
//   M = 16 output channels (o-tile), N = 16 positions p=(t,v), K = 64 in 16 steps.
// ---------------------------------------------------------------------------
__global__ __launch_bounds__(256) void stgcn_conv_kernel(
    const float* __restrict__ x2, const float* __restrict__ Wc,
    const float* __restrict__ bc, const float* __restrict__ xres,
    float* __restrict__ out)
{
  extern __shared__ float tile[];              // [64][18][24] = 110,592 B
  const int tid  = threadIdx.x;
  const int lane = tid & 31;
  const int wave = tid >> 5;
  const int hi   = (lane >> 4) & 1;
  const int l15  = lane & 15;
  const int n    = blockIdx.x >> 2;
  const int t0   = (blockIdx.x & 3) << 4;

  // warm residual lines for the epilogue (gfx1250 global_prefetch)
  {
    const float* rb = xres + (size_t)n * C_DIM * ROW_LEN + (size_t)t0 * V_DIM;
    for (int o = wave; o < C_DIM; o += 8)
      __builtin_prefetch(rb + (size_t)o * ROW_LEN + (lane << 4), 0, 1);
  }

  // ---- stage zero-padded x2 halo tile
  const float* x2n = x2 + (size_t)n * C_DIM * ROW_LEN;
  for (int idx = tid; idx < C_DIM * 18 * 24; idx += 256) {
    const int i   = idx / 432;
    const int rem = idx - i * 432;
    const int tt  = rem / 24;
    const int vv  = rem - tt * 24;
    const int gt  = t0 + tt - 1;
    const int gv  = vv - 1;
    float val = 0.f;
    if (gt >= 0 && gt < T_DIM && gv >= 0 && gv < V_DIM)
      val = x2n[(size_t)i * ROW_LEN + gt * V_DIM + gv];
    tile[idx] = val;
  }
  __syncthreads();

  // ---- jobs: (o-tile 0..3) x (p-tile 0..21); 16x22=352 positions in this t-block
  for (int j = wave; j < 4 * 22; j += 8) {
    const int ot = j & 3;
    const int pt = j >> 2;
    const int o0 = ot << 4;
    const int p  = (pt << 4) + l15;            // N index for this lane
    const int t  = p / V_DIM;                  // 0..15
    const int vv = p - t * V_DIM;              // 0..21

    v8f acc;                                   // seed with bias along M = r + 8*hi
#pragma unroll
    for (int r = 0; r < 8; ++r) acc[r] = bc[o0 + r + (hi << 3)];

    for (int dt = 0; dt < 3; ++dt) {
      for (int dv = 0; dv < 3; ++dv) {
        const int   bbase = (t + dt) * 24 + (vv + dv);     // halo-tile plane offset
        const float* wp   = Wc + (size_t)(o0 + l15) * (C_DIM * 9) + dt * 3 + dv;
#pragma unroll 4
        for (int k = 0; k < 16; ++k) {
          const int i0 = 4 * k + (hi ? 2 : 0);
          v2f a;                               // A 16x4: M = o0+l15, K = i0, i0+1
          a.x = wp[(size_t)i0 * 9];
          a.y = wp[(size_t)(i0 + 1) * 9];
          v2f b;                               // B 4x16: K = i0/i0+1, N = p
          b.x = tile[i0 * 432 + bbase];
          b.y = tile[(i0 + 1) * 432 + bbase];
          acc = __builtin_amdgcn_wmma_f32_16x16x4_f32(false, a, false, b, (short)0, acc, false, false);
        }
      }
    }

    // epilogue: residual add + store; M = r + 8*hi, position (t0+t, vv)
#pragma unroll
    for (int r = 0; r < 8; ++r) {
      const int o = o0 + r + (hi << 3);
      const size_t oidx = (((size_t)n * C_DIM + o) * T_DIM + (t0 + t)) * V_DIM + vv;
      out[oidx] = acc[r] + xres[oidx];
    }
  }
}

// ---------------------------------------------------------------------------
extern "C" void kernel_launch(void* const* d_in, const int* in_sizes, int n_in,
                              void* d_out, int out_size, void* d_ws, size_t ws_size,
                              hipStream_t stream) {
  const float* x    = (const float*)d_in[0];   // [256,64,64,22]
  const float* Tp   = (const float*)d_in[1];   // [22,64,64]
  const float* A    = (const float*)d_in[2];   // [64,22,22]
  const float* Afix = (const float*)d_in[3];   // [1,22,22]
  const float* Wc   = (const float*)d_in[4];   // [64,64,3,3]
  const float* bc   = (const float*)d_in[5];   // [64]
  float* out = (float*)d_out;
  float* ws  = (float*)d_ws;                   // x2 intermediate: 92.3 MB fp32

  const dim3 blk(256);
  const size_t lds1 = (size_t)V_DIM * NC_TILE * XS_STR * sizeof(float);   // 95,744 B
  stgcn_graph_kernel<<<dim3((N_DIM * C_DIM) / NC_TILE), blk, lds1, stream>>>(
      x, Tp, A, Afix, ws);

  const size_t lds2 = (size_t)C_DIM * 18 * 24 * sizeof(float);            // 110,592 B
  stgcn_conv_kernel<<<dim3(N_DIM * (T_DIM / 16)), blk, lds2, stream>>>(
      ws, Wc, bc, x, out);
}